// HybridModel_40862318854640
// MI455X (gfx1250) — compile-verified
//
#include <hip/hip_runtime.h>
#include <math.h>

#define N_LAYERS 10
#define PITCH 17  // LDS row pitch in floats; 17 is coprime with 64 banks -> conflict-free transposes

typedef __attribute__((ext_vector_type(2))) float v2f;
typedef __attribute__((ext_vector_type(8))) float v8f;

// ---------------------------------------------------------------------------
// Kernel 1: build per-layer 16x16 Kronecker unitaries U_A = U0(x)U1(x)U2(x)U3,
// U_B = U4(x)...(x)U7 from Rot angles, and store the four real A-operand
// matrices  A_r=[Re|-Im], A_i=[Im|Re]  (16x32 each) pre-swizzled into the
// exact WMMA 16x16x4 A-operand lane layout:
//   float index = l*2048 + mat*512 + tile*64 + lane*2 + p
//   holds element (row = lane&15, col = 4*tile + p + 2*(lane>>4))
// ---------------------------------------------------------------------------
__global__ __launch_bounds__(256) void build_unitaries(const float* __restrict__ w,
                                                       float* __restrict__ A)
{
  __shared__ float uaR[256], uaI[256], ubR[256], ubI[256];
  const int l = blockIdx.x;
  const int t = threadIdx.x;

  // All 8 Rot gates for this layer (redundant per thread; kernel is tiny).
  float gr[8][2][2], gi[8][2][2];
#pragma unroll
  for (int q = 0; q < 8; ++q) {
    float phi = w[l * 24 + q * 3 + 0];
    float th  = w[l * 24 + q * 3 + 1];
    float om  = w[l * 24 + q * 3 + 2];
    float sh, ch;
    sincosf(0.5f * th, &sh, &ch);
    float a = 0.5f * (phi + om), b = 0.5f * (phi - om);
    float ca, sa, cb, sb;
    sincosf(a, &sa, &ca);
    sincosf(b, &sb, &cb);
    // Rot = RZ(om) RY(th) RZ(phi):
    gr[q][0][0] =  ca * ch;  gi[q][0][0] = -sa * ch;   // e^{-i a} c
    gr[q][0][1] = -cb * sh;  gi[q][0][1] = -sb * sh;   // -e^{ i b} s
    gr[q][1][0] =  cb * sh;  gi[q][1][0] = -sb * sh;   // e^{-i b} s
    gr[q][1][1] =  ca * ch;  gi[q][1][1] =  sa * ch;   // e^{ i a} c
  }

  // Kronecker entry (r,c): wire0 is MSB of the nibble.
  {
    int r = t >> 4, c = t & 15;
    float ar = 1.f, ai = 0.f, br = 1.f, bi = 0.f;
#pragma unroll
    for (int j = 0; j < 4; ++j) {
      int rb = (r >> (3 - j)) & 1, cb = (c >> (3 - j)) & 1;
      float xr = gr[j][rb][cb], xi = gi[j][rb][cb];
      float nr = ar * xr - ai * xi, ni = ar * xi + ai * xr;
      ar = nr; ai = ni;
      xr = gr[4 + j][rb][cb]; xi = gi[4 + j][rb][cb];
      nr = br * xr - bi * xi; ni = br * xi + bi * xr;
      br = nr; bi = ni;
    }
    uaR[t] = ar; uaI[t] = ai; ubR[t] = br; ubI[t] = bi;
  }
  __syncthreads();

#pragma unroll
  for (int i = 0; i < 8; ++i) {
    int f    = t * 8 + i;           // 0..2047
    int mat  = f >> 9;
    int rem  = f & 511;
    int tile = rem >> 6;
    int lp   = rem & 63;
    int lane = lp >> 1, p = lp & 1;
    int row  = lane & 15;
    int col  = 4 * tile + p + 2 * (lane >> 4);
    int h    = col >> 4;
    int idx  = row * 16 + (col & 15);
    float val;
    if      (mat == 0) val = h ? -uaI[idx] : uaR[idx];  // A_r of U_A
    else if (mat == 1) val = h ?  uaR[idx] : uaI[idx];  // A_i of U_A
    else if (mat == 2) val = h ? -ubI[idx] : ubR[idx];  // A_r of U_B
    else               val = h ?  ubR[idx] : ubI[idx];  // A_i of U_B
    A[l * 2048 + f] = val;
  }
}

// ---------------------------------------------------------------------------
// Kernel 2: one batch element per wave32. State kept as stacked [Sr;Si]
// 32x16 in LDS; each layer = (U_A * S) via WMMA, transpose to LDS,
// (U_B * S^T) via WMMA, Gray-code CNOT permutation folded into the scatter.
// ---------------------------------------------------------------------------
__global__ __launch_bounds__(256) void qsim_main(const float* __restrict__ x,
                                                 const float* __restrict__ Am,
                                                 const float* __restrict__ W,
                                                 const float* __restrict__ bias,
                                                 float* __restrict__ out, int Bt)
{
  __shared__ float lds[8][2][32 * PITCH];
  const int wave = threadIdx.x >> 5;
  const int lane = threadIdx.x & 31;
  const int e    = blockIdx.x * 8 + wave;
  if (e >= Bt) return;                       // wave-uniform: EXEC stays all-1s
  float* buf0 = &lds[wave][0][0];
  float* buf1 = &lds[wave][1][0];
  const int hlf = lane >> 4;                 // lane half (WMMA layout)
  const int nl  = lane & 15;                 // N / row-within-half

  // Stage-2 scatter: D holds post-gate state at m=(N<<4)|M; CNOT chain sends it
  // to n = gray_decode(m)  (m = n ^ (n>>1)).  Layer-independent -> precompute.
  int scat[8], nidx[8];
#pragma unroll
  for (int v = 0; v < 8; ++v) {
    int m = (nl << 4) | (v + 8 * hlf);
    int n = m; n ^= n >> 4; n ^= n >> 2; n ^= n >> 1;
    nidx[v] = n;
    scat[v] = (n >> 4) * PITCH + (n & 15);
  }
  (void)nidx;

  // Amplitude embedding with normalize=True.
  float xs[8], ss = 0.f;
  const float* xp = x + (size_t)e * 256 + lane * 8;
#pragma unroll
  for (int i = 0; i < 8; ++i) { xs[i] = xp[i]; ss += xs[i] * xs[i]; }
#pragma unroll
  for (int d = 16; d > 0; d >>= 1) ss += __shfl_xor(ss, d, 32);
  float rn = rsqrtf(ss);
#pragma unroll
  for (int i = 0; i < 8; ++i) {
    int n = lane * 8 + i;
    buf0[(n >> 4) * PITCH + (n & 15)]        = xs[i] * rn;  // Sr
    buf0[(16 + (n >> 4)) * PITCH + (n & 15)] = 0.f;         // Si
  }
  asm volatile("s_wait_dscnt 0" ::: "memory");

  for (int l = 0; l < N_LAYERS; ++l) {
    const v2f* ab = (const v2f*)(Am + l * 2048);
    v2f a1r[8], a1i[8], a2r[8], a2i[8];
#pragma unroll
    for (int t = 0; t < 8; ++t) {            // coalesced, L2-resident
      a1r[t] = ab[      t * 32 + lane];
      a1i[t] = ab[256 + t * 32 + lane];
      a2r[t] = ab[512 + t * 32 + lane];
      a2i[t] = ab[768 + t * 32 + lane];
    }

    // Stage 1:  D = [Re|-Im ; Im|Re](U_A)  x  [Sr;Si]   (K=32 -> 8 WMMA each)
    v8f dr = {0,0,0,0,0,0,0,0}, di = {0,0,0,0,0,0,0,0};
#pragma unroll
    for (int t = 0; t < 8; ++t) {
      int k0 = 4 * t + 2 * hlf;
      v2f b;
      b.x = buf0[k0 * PITCH + nl];
      b.y = buf0[(k0 + 1) * PITCH + nl];
      dr = __builtin_amdgcn_wmma_f32_16x16x4_f32(false, a1r[t], false, b, (short)0, dr, false, false);
      di = __builtin_amdgcn_wmma_f32_16x16x4_f32(false, a1i[t], false, b, (short)0, di, false, false);
    }
    // Write T1^T into buf1 as stacked [T1r^T ; T1i^T].
#pragma unroll
    for (int v = 0; v < 8; ++v) {
      int M = v + 8 * hlf;
      buf1[nl * PITCH + M]        = dr[v];
      buf1[(16 + nl) * PITCH + M] = di[v];
    }
    asm volatile("s_wait_dscnt 0" ::: "memory");

    // Stage 2:  E = U_B x T1^T   (same structure)
    v8f er = {0,0,0,0,0,0,0,0}, ei = {0,0,0,0,0,0,0,0};
#pragma unroll
    for (int t = 0; t < 8; ++t) {
      int k0 = 4 * t + 2 * hlf;
      v2f b;
      b.x = buf1[k0 * PITCH + nl];
      b.y = buf1[(k0 + 1) * PITCH + nl];
      er = __builtin_amdgcn_wmma_f32_16x16x4_f32(false, a2r[t], false, b, (short)0, er, false, false);
      ei = __builtin_amdgcn_wmma_f32_16x16x4_f32(false, a2i[t], false, b, (short)0, ei, false, false);
    }
    // Transpose-back + CNOT permutation in one scatter.
#pragma unroll
    for (int v = 0; v < 8; ++v) {
      buf0[scat[v]]             = er[v];
      buf0[scat[v] + 16 * PITCH] = ei[v];
    }
    asm volatile("s_wait_dscnt 0" ::: "memory");
  }

  // <Z_q> then tiny linear head.
  float z[8] = {0,0,0,0,0,0,0,0};
#pragma unroll
  for (int i = 0; i < 8; ++i) {
    int n = lane * 8 + i;
    float re = buf0[(n >> 4) * PITCH + (n & 15)];
    float im = buf0[(16 + (n >> 4)) * PITCH + (n & 15)];
    float p  = re * re + im * im;
#pragma unroll
    for (int q = 0; q < 8; ++q)
      z[q] += ((n >> (7 - q)) & 1) ? -p : p;
  }
#pragma unroll
  for (int d = 16; d > 0; d >>= 1) {
#pragma unroll
    for (int q = 0; q < 8; ++q) z[q] += __shfl_xor(z[q], d, 32);
  }
  if (lane < 10) {
    float acc = bias[lane];
#pragma unroll
    for (int q = 0; q < 8; ++q) acc += z[q] * W[lane * 8 + q];
    out[(size_t)e * 10 + lane] = acc;
  }
}

// ---------------------------------------------------------------------------
extern "C" void kernel_launch(void* const* d_in, const int* in_sizes, int n_in,
                              void* d_out, int out_size, void* d_ws, size_t ws_size,
                              hipStream_t stream) {
  const float* x    = (const float*)d_in[0];   // [B,256]
  const float* w    = (const float*)d_in[1];   // [10,8,3]
  const float* W    = (const float*)d_in[2];   // [10,8]
  const float* bias = (const float*)d_in[3];   // [10]
  float* Amats      = (float*)d_ws;            // 10*2048 floats = 80 KB
  int Bt = in_sizes[0] / 256;

  build_unitaries<<<N_LAYERS, 256, 0, stream>>>(w, Amats);
  qsim_main<<<(Bt + 7) / 8, 256, 0, stream>>>(x, Amats, W, bias, (float*)d_out, Bt);
}